// MeshRender_88450556494137
// MI455X (gfx1250) — compile-verified
//
#include <hip/hip_runtime.h>
#include <math.h>

#define RES_W 512
#define N_FACES 128
#define MAXD 100.0f
#define REPS 1e-8f
#define N_PIX (RES_W * RES_W)

typedef float v2f __attribute__((ext_vector_type(2)));
typedef float v8f __attribute__((ext_vector_type(8)));
typedef int   v4i_ls __attribute__((vector_size(16)));   // matches builtin param type

// workspace float layout
#define WS_B    0     // 8 groups * 4 functionals * 4 krows * 16 cols = 2048 floats
#define WS_INVA 2048  // 128 floats (1/area, 0 if |area|<=EPS)  -- contiguous with WS_B
#define WS_FACE 2176  // 128 * 8 floats: x0,y0,x1,y1,x2,y2,pad,pad
#define WS_RASTER_FLOATS 2176   // B tiles + invA staged to LDS as one block

#if defined(__has_builtin)
#if __has_builtin(__builtin_amdgcn_global_load_async_to_lds_b128) && \
    __has_builtin(__builtin_amdgcn_s_wait_asynccnt)
#define MESH_ASYNC_LDS 1
#endif
#endif

__device__ __forceinline__ void edge_coef(float ax, float ay, float bx, float by,
                                          float& Ax, float& Ay, float& C) {
    // edge(a,b,p) = (bx-ax)*(py-ay) - (by-ay)*(px-ax) = Ax*px + Ay*py + C
    Ax = -(by - ay);
    Ay = (bx - ax);
    C  = (by - ay) * ax - (bx - ax) * ay;
}

__global__ __launch_bounds__(128) void mesh_setup_kernel(
    const float* __restrict__ verts, const int* __restrict__ faces,
    const float* __restrict__ Ks, const float* __restrict__ RTs,
    float* __restrict__ wsf)
{
    int f = threadIdx.x;
    if (f >= N_FACES) return;

    float r00 = RTs[0], r01 = RTs[1], r02 = RTs[2],  r03 = RTs[3];
    float r10 = RTs[4], r11 = RTs[5], r12 = RTs[6],  r13 = RTs[7];
    float r20 = RTs[8], r21 = RTs[9], r22 = RTs[10], r23 = RTs[11];
    float k00 = Ks[0], k02 = Ks[2], k11 = Ks[4], k12 = Ks[5];

    float X[3], Y[3], Z[3];
    #pragma unroll
    for (int k = 0; k < 3; k++) {
        int vi = faces[f * 3 + k];
        float vx = verts[vi * 3 + 0], vy = verts[vi * 3 + 1], vz = verts[vi * 3 + 2];
        float cx = r00 * vx + r01 * vy + r02 * vz + r03;
        float cy = r10 * vx + r11 * vy + r12 * vz + r13;
        float cz = r20 * vx + r21 * vy + r22 * vz + r23;
        X[k] = k00 * cx + k02;
        Y[k] = k11 * cy + k12;
        Z[k] = cz;
    }

    float Ax[3], Ay[3], C[3];
    edge_coef(X[1], Y[1], X[2], Y[2], Ax[0], Ay[0], C[0]); // w0 = edge(v1,v2,p)
    edge_coef(X[2], Y[2], X[0], Y[0], Ax[1], Ay[1], C[1]); // w1 = edge(v2,v0,p)
    edge_coef(X[0], Y[0], X[1], Y[1], Ax[2], Ay[2], C[2]); // w2 = edge(v0,v1,p)

    // 2*signed area (== w0+w1+w2, constant over p)
    float area = (X[1] - X[0]) * (Y[2] - Y[0]) - (Y[1] - Y[0]) * (X[2] - X[0]);
    float inva = (fabsf(area) > REPS) ? (1.0f / area) : 0.0f; // 0 folds |area|<=EPS invalidation

    // z*area plane: w0*z0 + w1*z1 + w2*z2, also affine in (px,py)
    float Axz = Ax[0] * Z[0] + Ax[1] * Z[1] + Ax[2] * Z[2];
    float Ayz = Ay[0] * Z[0] + Ay[1] * Z[1] + Ay[2] * Z[2];
    float Cz  = C[0]  * Z[0] + C[1]  * Z[1] + C[2]  * Z[2];

    int g = f >> 4;   // face group of 16
    int n = f & 15;   // column within group
    float* tg = wsf + WS_B + g * 256;
    #pragma unroll
    for (int q = 0; q < 3; q++) {
        tg[(q * 4 + 0) * 16 + n] = Ax[q];
        tg[(q * 4 + 1) * 16 + n] = Ay[q];
        tg[(q * 4 + 2) * 16 + n] = C[q];
        tg[(q * 4 + 3) * 16 + n] = 0.0f;
    }
    tg[(3 * 4 + 0) * 16 + n] = Axz;
    tg[(3 * 4 + 1) * 16 + n] = Ayz;
    tg[(3 * 4 + 2) * 16 + n] = Cz;
    tg[(3 * 4 + 3) * 16 + n] = 0.0f;

    wsf[WS_INVA + f] = inva;

    float* fd = wsf + WS_FACE + f * 8;
    fd[0] = X[0]; fd[1] = Y[0];
    fd[2] = X[1]; fd[3] = Y[1];
    fd[4] = X[2]; fd[5] = Y[2];
    fd[6] = 0.0f; fd[7] = 0.0f;
}

// One wave = 16 consecutive pixels in a row. 32 tiles per image row.
// Per group of 16 faces: 4 WMMA f32 16x16x4 ops (w0,w1,w2, z*area).
__global__ __launch_bounds__(256) void mesh_raster_kernel(
    const float* __restrict__ wsf,
    float* __restrict__ depth_out, int* __restrict__ win_out)
{
    __shared__ __align__(16) float sB[WS_RASTER_FLOATS]; // 2048 B-tile floats + 128 invA

    // Stage B tiles + invA (8.7 KB) into LDS once per workgroup (async-to-LDS path).
#if MESH_ASYNC_LDS
    for (int i = threadIdx.x; i < WS_RASTER_FLOATS / 4; i += 256) {
        __builtin_amdgcn_global_load_async_to_lds_b128(
            (__attribute__((address_space(1))) v4i_ls*)(wsf + i * 4),
            (__attribute__((address_space(3))) v4i_ls*)(sB + i * 4),
            0, 0);
    }
    __builtin_amdgcn_s_wait_asynccnt(0);
#else
    for (int i = threadIdx.x; i < WS_RASTER_FLOATS / 4; i += 256) {
        const float4 t = ((const float4*)wsf)[i];
        ((float4*)sB)[i] = t;
    }
#endif
    __syncthreads();

    const int lane = threadIdx.x & 31;
    const int wave = threadIdx.x >> 5;
    const int tile = blockIdx.x * 8 + wave;   // 16384 tiles total
    const int py   = tile >> 5;
    const int px0  = (tile & 31) << 4;
    const int n    = lane & 15;
    const bool hi  = lane >= 16;

    // A (16x4 f32): lanes 0-15 hold K0/K1 = (px, py); lanes 16-31 hold K2/K3 = (1, 0)
    v2f a;
    a.x = hi ? 1.0f : ((float)(px0 + n) + 0.5f);
    a.y = hi ? 0.0f : ((float)py + 0.5f);

    float bestz[8];
    int   bestf[8];
    #pragma unroll
    for (int r = 0; r < 8; r++) { bestz[r] = MAXD; bestf[r] = 0; }

    const int k0 = hi ? 2 : 0;
    // unroll-by-2: enough to overlap ds_loads of group g+1 with WMMAs of group g,
    // while staying inside the direct 256-VGPR window (avoids s_set_vgpr_msb churn).
    #pragma unroll 2
    for (int g = 0; g < 8; g++) {
        const float* tg = sB + g * 256;
        v2f b0, b1, b2, b3;
        b0.x = tg[(0 * 4 + k0) * 16 + n];  b0.y = tg[(0 * 4 + k0 + 1) * 16 + n];
        b1.x = tg[(1 * 4 + k0) * 16 + n];  b1.y = tg[(1 * 4 + k0 + 1) * 16 + n];
        b2.x = tg[(2 * 4 + k0) * 16 + n];  b2.y = tg[(2 * 4 + k0 + 1) * 16 + n];
        b3.x = tg[(3 * 4 + k0) * 16 + n];  b3.y = tg[(3 * 4 + k0 + 1) * 16 + n];

        v8f c0 = {}, c1 = {}, c2 = {}, c3 = {};
        c0 = __builtin_amdgcn_wmma_f32_16x16x4_f32(false, a, false, b0, (short)0, c0, false, false);
        c1 = __builtin_amdgcn_wmma_f32_16x16x4_f32(false, a, false, b1, (short)0, c1, false, false);
        c2 = __builtin_amdgcn_wmma_f32_16x16x4_f32(false, a, false, b2, (short)0, c2, false, false);
        c3 = __builtin_amdgcn_wmma_f32_16x16x4_f32(false, a, false, b3, (short)0, c3, false, false);

        const int face = g * 16 + n;
        const float ia = sB[WS_INVA + face];

        // D layout: VGPR r, lanes 0-15 -> pixel M=r, lanes 16-31 -> pixel M=r+8; N = lane&15 = face col
        #pragma unroll
        for (int r = 0; r < 8; r++) {
            float w0 = c0[r], w1 = c1[r], w2 = c2[r], za = c3[r];
            // all >=0 or all <=0  <=>  min3 >= 0 || max3 <= 0 (v_min3/v_max3_num_f32)
            float mn = fminf(fminf(w0, w1), w2);
            float mx = fmaxf(fmaxf(w0, w1), w2);
            float zp = za * ia;   // ia==0 => zp==0 => rejected (folds |area|<=EPS)
            bool accept = ((mn >= 0.0f) || (mx <= 0.0f)) && (zp > 0.0f) && (zp < bestz[r]);
            bestz[r] = accept ? zp   : bestz[r];
            bestf[r] = accept ? face : bestf[r];
        }
    }

    // min/argmin across the 16 face columns (stays within each 16-lane half)
    #pragma unroll
    for (int r = 0; r < 8; r++) {
        float z = bestz[r];
        int   f = bestf[r];
        #pragma unroll
        for (int s = 1; s < 16; s <<= 1) {
            float oz = __shfl_xor(z, s, 32);
            int   of = __shfl_xor(f, s, 32);
            if (oz < z || (oz == z && of < f)) { z = oz; f = of; }
        }
        bestz[r] = z; bestf[r] = f;
    }

    if (n == 0) { // lane 0 stores pixels 0..7, lane 16 stores pixels 8..15
        int base = py * RES_W + px0 + (hi ? 8 : 0);
        #pragma unroll
        for (int r = 0; r < 8; r++) {
            depth_out[base + r] = bestz[r];
            win_out[base + r]   = bestf[r];
        }
    }
}

__global__ __launch_bounds__(256) void mesh_shade_kernel(
    const float* __restrict__ normals, const float* __restrict__ uvs,
    const float* __restrict__ tex,
    const float* __restrict__ view_dir, const float* __restrict__ light_dir,
    const float* __restrict__ wsf, const float* __restrict__ depth,
    const int* winIn, float* __restrict__ rgb_out, int* mask_out)
{
    int pix = blockIdx.x * blockDim.x + threadIdx.x;
    if (pix >= N_PIX) return;
    int py = pix >> 9, px = pix & 511;

    float d = depth[pix];
    int   w = winIn[pix];
    bool hit = d < MAXD;

    const float* fd = wsf + WS_FACE + w * 8;
    float x0 = fd[0], y0 = fd[1], x1 = fd[2], y1 = fd[3], x2 = fd[4], y2 = fd[5];
    float ppx = (float)px + 0.5f, ppy = (float)py + 0.5f;

    float u0 = (x2 - x1) * (ppy - y1) - (y2 - y1) * (ppx - x1);
    float u1 = (x0 - x2) * (ppy - y2) - (y0 - y2) * (ppx - x2);
    float u2 = (x1 - x0) * (ppy - y0) - (y1 - y0) * (ppx - x0);
    float ar = u0 + u1 + u2;
    float inv2 = (fabsf(ar) > REPS) ? (1.0f / ar) : 1.0f;
    float b0 = u0 * inv2, b1 = u1 * inv2, b2 = u2 * inv2;

    const float* nw = normals + w * 9;
    float nx = b0 * nw[0] + b1 * nw[3] + b2 * nw[6];
    float ny = b0 * nw[1] + b1 * nw[4] + b2 * nw[7];
    float nz = b0 * nw[2] + b1 * nw[5] + b2 * nw[8];
    float nl = sqrtf(nx * nx + ny * ny + nz * nz) + REPS;
    nx /= nl; ny /= nl; nz /= nl;

    const float* uvw = uvs + w * 6;
    float u = b0 * uvw[0] + b1 * uvw[2] + b2 * uvw[4];
    float v = b0 * uvw[1] + b1 * uvw[3] + b2 * uvw[5];

    float uu = fminf(fmaxf(u, 0.0f), 1.0f) * 511.0f;
    float vv = fminf(fmaxf(v, 0.0f), 1.0f) * 511.0f;
    int ix0 = (int)floorf(uu), iy0 = (int)floorf(vv);
    int ix1 = min(ix0 + 1, 511), iy1 = min(iy0 + 1, 511);
    float fx = uu - (float)ix0, fy = vv - (float)iy0;

    const float* t00 = tex + (iy0 * 512 + ix0) * 3;
    const float* t01 = tex + (iy0 * 512 + ix1) * 3;
    const float* t10 = tex + (iy1 * 512 + ix0) * 3;
    const float* t11 = tex + (iy1 * 512 + ix1) * 3;
    float w00 = (1.0f - fx) * (1.0f - fy), w01 = fx * (1.0f - fy);
    float w10 = (1.0f - fx) * fy,          w11 = fx * fy;

    float lx = light_dir[0], ly = light_dir[1], lz = light_dir[2];
    float ll = sqrtf(lx * lx + ly * ly + lz * lz) + REPS;
    lx /= ll; ly /= ll; lz /= ll;
    float vx = view_dir[0], vy = view_dir[1], vz = view_dir[2];
    float vl = sqrtf(vx * vx + vy * vy + vz * vz) + REPS;
    vx /= vl; vy /= vl; vz /= vl;
    float hx = lx + vx, hy = ly + vy, hz = lz + vz;
    float hl = sqrtf(hx * hx + hy * hy + hz * hz) + REPS;
    hx /= hl; hy /= hl; hz /= hl;

    float diff = fmaxf(nx * lx + ny * ly + nz * lz, 0.0f);
    float sp   = fmaxf(nx * hx + ny * hy + nz * hz, 0.0f);
    float sp2 = sp * sp, sp4 = sp2 * sp2, sp8 = sp4 * sp4, sp16 = sp8 * sp8;
    float spec  = 0.8f * sp16;
    float shade = 0.3f + 0.8f * diff;

    #pragma unroll
    for (int c = 0; c < 3; c++) {
        float col = t00[c] * w00 + t01[c] * w01 + t10[c] * w10 + t11[c] * w11;
        rgb_out[pix * 3 + c] = hit ? (col * shade + spec) : 0.0f;
    }
    mask_out[pix] = hit ? 1 : 0;
}

extern "C" void kernel_launch(void* const* d_in, const int* in_sizes, int n_in,
                              void* d_out, int out_size, void* d_ws, size_t ws_size,
                              hipStream_t stream) {
    const float* verts   = (const float*)d_in[0];
    const int*   faces   = (const int*)d_in[1];
    const float* normals = (const float*)d_in[2];
    const float* uvs     = (const float*)d_in[3];
    const float* tex     = (const float*)d_in[4];
    const float* Ks      = (const float*)d_in[5];
    const float* RTs     = (const float*)d_in[6];
    const float* view    = (const float*)d_in[7];
    const float* light   = (const float*)d_in[8];

    float* depth_out = (float*)d_out;                 // N_PIX f32
    float* rgb_out   = depth_out + N_PIX;             // N_PIX*3 f32
    int*   mask_out  = (int*)(depth_out + N_PIX * 4); // N_PIX i32; doubles as win-index scratch
    float* wsf       = (float*)d_ws;

    mesh_setup_kernel<<<1, 128, 0, stream>>>(verts, faces, Ks, RTs, wsf);
    mesh_raster_kernel<<<(N_PIX / 16) / 8, 256, 0, stream>>>(wsf, depth_out, mask_out);
    mesh_shade_kernel<<<N_PIX / 256, 256, 0, stream>>>(normals, uvs, tex, view, light,
                                                       wsf, depth_out, mask_out,
                                                       rgb_out, mask_out);
}